// _NGCF_52175262712265
// MI455X (gfx1250) — compile-verified
//
#include <hip/hip_runtime.h>
#include <hip/hip_bf16.h>
#include <math.h>

#define N_USERS 100000
#define N_ITEMS 50000
#define N_NODES (N_USERS + N_ITEMS)
#define NNZ_    2400000
#define EMB     64
#define NLAYERS 3
#define BATCH_  16384
#define REG_    1e-05f

typedef float v2f __attribute__((ext_vector_type(2)));
typedef float v8f __attribute__((ext_vector_type(8)));

// ---------------------------------------------------------------- utilities
__global__ void k_zero_f4(float4* __restrict__ p, long long n4) {
    long long t = (long long)blockIdx.x * blockDim.x + threadIdx.x;
    if (t < n4) p[t] = make_float4(0.f, 0.f, 0.f, 0.f);
}

__global__ void k_concat(const float* __restrict__ ue, const float* __restrict__ ie,
                         float4* __restrict__ ego4) {
    long long t = (long long)blockIdx.x * blockDim.x + threadIdx.x;
    long long n4 = (long long)N_NODES * (EMB / 4);
    if (t >= n4) return;
    long long userEnd = (long long)N_USERS * (EMB / 4);
    if (t < userEnd)
        ego4[t] = ((const float4*)ue)[t];
    else
        ego4[t] = ((const float4*)ie)[t - userEnd];
}

// ------------------------------------------------------------------- SpMM
// one wave per edge; lane handles dims {2l, 2l+1}; hardware f32 atomics
__global__ void k_spmm(const int* __restrict__ row, const int* __restrict__ col,
                       const float* __restrict__ val, const float* __restrict__ x,
                       float* __restrict__ y) {
    long long t = (long long)blockIdx.x * blockDim.x + threadIdx.x;
    int e = (int)(t >> 5);
    int l = (int)(t & 31);
    if (e >= NNZ_) return;
    int c = col[e];
    int r = row[e];
    float v = val[e];
    const float2 xv = *(const float2*)(x + (long long)c * EMB + 2 * l);
    float* dst = y + (long long)r * EMB + 2 * l;
    unsafeAtomicAdd(dst + 0, v * xv.x);
    unsafeAtomicAdd(dst + 1, v * xv.y);
}

// ------------------------------------------------------- dense layer (WMMA)
// ego_new = leaky_relu( (ego+side)@W1 + (ego*side)@W2 + b1 + b2 ), in place.
// Each wave owns a fixed output-column half (32 cols = 2 n-tiles): W fragments
// for that half live in 128 VGPRs, loaded once per kernel. A fragments are
// software-pipelined one k-step ahead to overlap L2 latency with the WMMAs.
__global__ void __launch_bounds__(128) k_dense(
        const float* __restrict__ ego, const float* __restrict__ side,
        float* __restrict__ ego_out,
        const float* __restrict__ W1, const float* __restrict__ b1,
        const float* __restrict__ W2, const float* __restrict__ b2) {
    __shared__ float WT1[EMB * EMB];   // transposed: WT[n*64 + k]
    __shared__ float WT2[EMB * EMB];
    __shared__ float bs[EMB];

    for (int idx = threadIdx.x; idx < EMB * EMB; idx += blockDim.x) {
        int r = idx >> 6, c = idx & 63;
        WT1[c * EMB + r] = W1[idx];
        WT2[c * EMB + r] = W2[idx];
    }
    if (threadIdx.x < EMB) bs[threadIdx.x] = b1[threadIdx.x] + b2[threadIdx.x];
    __syncthreads();

    const int lane = threadIdx.x & 31;
    const int half = lane >> 4;         // which 16-lane half (K pair select)
    const int ln   = lane & 15;
    const int waveInBlock   = threadIdx.x >> 5;
    const int wavesPerBlock = blockDim.x >> 5;
    const int waveGlobal    = blockIdx.x * wavesPerBlock + waveInBlock;
    const int nWaves        = gridDim.x * wavesPerBlock;   // even by construction

    const int nh         = waveGlobal & 1;        // fixed column half for this wave
    const int tile0      = waveGlobal >> 1;
    const int tileStride = nWaves >> 1;
    const int nTiles     = N_NODES / 16;          // 9375 exactly

    // ---- preload this wave's B fragments into registers (once per kernel)
    // B fragment (4x16): n = l%16 (+16*nt), K = 4*kb + 2*half + v
    v2f B1[2][16];
    v2f B2[2][16];
    float bv[2];
    #pragma unroll
    for (int t = 0; t < 2; ++t) {
        const int n = (2 * nh + t) * 16 + ln;
        bv[t] = bs[n];
        #pragma unroll
        for (int kb = 0; kb < 16; ++kb) {
            const int kOff = kb * 4 + half * 2;
            B1[t][kb] = *(const v2f*)(&WT1[n * EMB + kOff]);
            B2[t][kb] = *(const v2f*)(&WT2[n * EMB + kOff]);
        }
    }

    for (int tile = tile0; tile < nTiles; tile += tileStride) {
        const int node0 = tile * 16;
        const float* egoRow  = ego  + (long long)(node0 + ln) * EMB;
        const float* sideRow = side + (long long)(node0 + ln) * EMB;

        v8f acc[2];
        #pragma unroll
        for (int t = 0; t < 2; ++t)
            #pragma unroll
            for (int r = 0; r < 8; ++r) acc[t][r] = bv[t];

        // A fragment (f32 16x4): lane m = l%16; VGPR v holds K = 4*kb + 2*half + v
        // Software pipeline: fragments for step kb+1 are in flight while the
        // four WMMAs of step kb execute.
        v2f e2 = *(const v2f*)(egoRow + half * 2);
        v2f s2 = *(const v2f*)(sideRow + half * 2);
        #pragma unroll
        for (int kb = 0; kb < 16; ++kb) {
            v2f eN = {0.f, 0.f}, sN = {0.f, 0.f};
            if (kb < 15) {
                const int kOffN = (kb + 1) * 4 + half * 2;
                eN = *(const v2f*)(egoRow + kOffN);
                sN = *(const v2f*)(sideRow + kOffN);
            }
            v2f a1 = e2 + s2;     // side_LI
            v2f a2 = e2 * s2;     // ego * side_L
            #pragma unroll
            for (int t = 0; t < 2; ++t) {
                acc[t] = __builtin_amdgcn_wmma_f32_16x16x4_f32(
                    false, a1, false, B1[t][kb], (short)0, acc[t], false, false);
                acc[t] = __builtin_amdgcn_wmma_f32_16x16x4_f32(
                    false, a2, false, B2[t][kb], (short)0, acc[t], false, false);
            }
            e2 = eN;
            s2 = sN;
        }

        // leaky_relu(0.01) + store. D layout: VGPR r -> row r + 8*half, col = l%16
        #pragma unroll
        for (int t = 0; t < 2; ++t) {
            #pragma unroll
            for (int r = 0; r < 8; ++r) {
                float v = acc[t][r];
                v = v > 0.f ? v : 0.01f * v;
                int mm = r + 8 * half;
                ego_out[(long long)(node0 + mm) * EMB + (2 * nh + t) * 16 + ln] = v;
            }
        }
    }
}

// ----------------------------------------------- gather + dot accumulation
__global__ void k_gather(const float* __restrict__ emb,
                         const int* __restrict__ u, const int* __restrict__ i,
                         const int* __restrict__ j,
                         float* __restrict__ y_ui, float* __restrict__ y_uj,
                         float* __restrict__ l2, int normalize) {
    int wave = (blockIdx.x * blockDim.x + threadIdx.x) >> 5;
    int lane = threadIdx.x & 31;
    if (wave >= BATCH_) return;

    int un = u[wave];
    int pn = N_USERS + i[wave];
    int qn = N_USERS + j[wave];

    float2 uu = *(const float2*)(emb + (long long)un * EMB + 2 * lane);
    float2 pp = *(const float2*)(emb + (long long)pn * EMB + 2 * lane);
    float2 qq = *(const float2*)(emb + (long long)qn * EMB + 2 * lane);

    float su = uu.x * uu.x + uu.y * uu.y;
    float sp = pp.x * pp.x + pp.y * pp.y;
    float sq = qq.x * qq.x + qq.y * qq.y;
    #pragma unroll
    for (int off = 16; off > 0; off >>= 1) {
        su += __shfl_xor(su, off, 32);
        sp += __shfl_xor(sp, off, 32);
        sq += __shfl_xor(sq, off, 32);
    }
    if (normalize) {
        float iu = 1.f / fmaxf(sqrtf(su), 1e-12f);
        float ip = 1.f / fmaxf(sqrtf(sp), 1e-12f);
        float iq = 1.f / fmaxf(sqrtf(sq), 1e-12f);
        uu.x *= iu; uu.y *= iu;
        pp.x *= ip; pp.y *= ip;
        qq.x *= iq; qq.y *= iq;
        su *= iu * iu; sp *= ip * ip; sq *= iq * iq;
    }
    float dui = uu.x * pp.x + uu.y * pp.y;
    float duj = uu.x * qq.x + uu.y * qq.y;
    #pragma unroll
    for (int off = 16; off > 0; off >>= 1) {
        dui += __shfl_xor(dui, off, 32);
        duj += __shfl_xor(duj, off, 32);
    }
    if (lane == 0) {
        y_ui[wave] += dui;
        y_uj[wave] += duj;
        l2[wave]   += 0.5f * (su + sp + sq);
    }
}

// ----------------------------------------------------------- final reduce
__global__ void k_loss(const float* __restrict__ y_ui, const float* __restrict__ y_uj,
                       const float* __restrict__ l2, float* __restrict__ out) {
    __shared__ float sl[256];
    __shared__ float sr[256];
    float a = 0.f, b = 0.f;
    for (int t = threadIdx.x; t < BATCH_; t += blockDim.x) {
        float z = y_ui[t] - y_uj[t];
        a += fmaxf(-z, 0.f) + log1pf(expf(-fabsf(z)));   // -log_sigmoid(z)
        b += l2[t];
    }
    sl[threadIdx.x] = a;
    sr[threadIdx.x] = b;
    __syncthreads();
    for (int s = 128; s > 0; s >>= 1) {
        if (threadIdx.x < s) {
            sl[threadIdx.x] += sl[threadIdx.x + s];
            sr[threadIdx.x] += sr[threadIdx.x + s];
        }
        __syncthreads();
    }
    if (threadIdx.x == 0)
        out[0] = sl[0] / (float)BATCH_ + REG_ * (sr[0] / (float)BATCH_);
}

// ---------------------------------------------------------------- launch
extern "C" void kernel_launch(void* const* d_in, const int* in_sizes, int n_in,
                              void* d_out, int out_size, void* d_ws, size_t ws_size,
                              hipStream_t stream) {
    const int*   adj_row  = (const int*)d_in[0];
    const int*   adj_col  = (const int*)d_in[1];
    const float* adj_val  = (const float*)d_in[2];
    const float* user_emb = (const float*)d_in[3];
    const float* item_emb = (const float*)d_in[4];
    const float* W1       = (const float*)d_in[5];
    const float* b1       = (const float*)d_in[6];
    const float* W2       = (const float*)d_in[7];
    const float* b2       = (const float*)d_in[8];
    const int*   u        = (const int*)d_in[9];
    const int*   ii       = (const int*)d_in[10];
    const int*   jj       = (const int*)d_in[11];

    const long long embElems = (long long)N_NODES * EMB;   // 9.6M floats
    float* ego  = (float*)d_ws;
    float* side = ego + embElems;
    float* y_ui = side + embElems;
    float* y_uj = y_ui + BATCH_;
    float* l2   = y_uj + BATCH_;

    // ego = concat(user_emb, item_emb)
    {
        long long n4 = embElems / 4;
        int blk = 256;
        k_concat<<<(int)((n4 + blk - 1) / blk), blk, 0, stream>>>(
            user_emb, item_emb, (float4*)ego);
    }
    // zero per-batch accumulators (y_ui, y_uj, l2 contiguous)
    {
        long long n4 = (3LL * BATCH_) / 4;
        int blk = 256;
        k_zero_f4<<<(int)((n4 + blk - 1) / blk), blk, 0, stream>>>((float4*)y_ui, n4);
    }
    // layer-0 contribution (raw ego, no normalization)
    {
        int blk = 256;
        long long thr = (long long)BATCH_ * 32;
        k_gather<<<(int)((thr + blk - 1) / blk), blk, 0, stream>>>(
            ego, u, ii, jj, y_ui, y_uj, l2, 0);
    }

    for (int k = 0; k < NLAYERS; ++k) {
        // side = 0
        {
            long long n4 = embElems / 4;
            int blk = 256;
            k_zero_f4<<<(int)((n4 + blk - 1) / blk), blk, 0, stream>>>((float4*)side, n4);
        }
        // side = A @ ego (scatter-add over edges)
        {
            int blk = 256;
            long long thr = (long long)NNZ_ * 32;
            k_spmm<<<(int)((thr + blk - 1) / blk), blk, 0, stream>>>(
                adj_row, adj_col, adj_val, ego, side);
        }
        // ego = leaky_relu((ego+side)@W1 + (ego*side)@W2 + b1 + b2)  [WMMA]
        {
            int blk = 128;      // 4 waves/block; wave = (tile-stream, column half)
            int grd = 512;      // 2048 waves -> 1024 tile streams over 9375 tiles
            k_dense<<<grd, blk, 0, stream>>>(ego, side, ego,
                                             W1 + (long long)k * EMB * EMB,
                                             b1 + (long long)k * EMB,
                                             W2 + (long long)k * EMB * EMB,
                                             b2 + (long long)k * EMB);
        }
        // accumulate this layer's (normalized) contribution
        {
            int blk = 256;
            long long thr = (long long)BATCH_ * 32;
            k_gather<<<(int)((thr + blk - 1) / blk), blk, 0, stream>>>(
                ego, u, ii, jj, y_ui, y_uj, l2, 1);
        }
    }

    k_loss<<<1, 256, 0, stream>>>(y_ui, y_uj, l2, (float*)d_out);
}